// GATEncoder_67413806678215
// MI455X (gfx1250) — compile-verified
//
#include <hip/hip_runtime.h>
#include <hip/hip_bf16.h>

#define HEADS 4
#define CPH   64
#define HC    256
#define SLOPE 0.2f

typedef __attribute__((ext_vector_type(2))) float v2f;
typedef __attribute__((ext_vector_type(8))) float v8f;

// ---------------------------------------------------------------------------
// GEMM: H[N,256] = X[N,FIN] @ W[FIN,256] using V_WMMA_F32_16X16X4_F32.
// One block = 256 threads = 8 waves, covers a 64-row strip (4 row-tiles).
// Each wave owns two 16-col panels (cols wave*16 and (wave+8)*16); per k-step
// one B fragment is loaded and reused by 4 independent WMMAs (one per
// row-tile), cutting W L2 traffic 4x vs a 16-row strip. X strip staged in
// padded LDS (+1 float kills the 16-way bank conflict on column reads).
// ---------------------------------------------------------------------------
template <int FIN>
__global__ __launch_bounds__(256) void gemm_wmma_f32(
    const float* __restrict__ X, const float* __restrict__ W,
    float* __restrict__ H, int nrows) {
  __shared__ float xs[64][FIN + 1];
  const int row0 = blockIdx.x * 64;
  for (int i = threadIdx.x; i < 64 * FIN; i += 256) {
    const int r = i / FIN, c = i % FIN;
    int rr = row0 + r;
    if (rr >= nrows) rr = nrows - 1;  // clamp; stores are guarded below
    xs[r][c] = X[(size_t)rr * FIN + c];
  }
  __syncthreads();

  const int wave = threadIdx.x >> 5;
  const int lane = threadIdx.x & 31;
  const int m    = lane & 15;
  const int hi   = lane >> 4;  // 0: holds K=k,k+1 ; 1: holds K=k+2,k+3

#pragma unroll
  for (int t = 0; t < 2; ++t) {
    const int c0 = (wave + t * 8) * 16;
    v8f acc0 = {0.f, 0.f, 0.f, 0.f, 0.f, 0.f, 0.f, 0.f};
    v8f acc1 = acc0, acc2 = acc0, acc3 = acc0;
    for (int k = 0; k < FIN; k += 4) {
      const int kk = k + (hi << 1);
      v2f B;
      B.x = W[(size_t)kk * HC + c0 + m];
      B.y = W[(size_t)(kk + 1) * HC + c0 + m];
      v2f A0, A1, A2, A3;
      A0.x = xs[m][kk];       A0.y = xs[m][kk + 1];
      A1.x = xs[m + 16][kk];  A1.y = xs[m + 16][kk + 1];
      A2.x = xs[m + 32][kk];  A2.y = xs[m + 32][kk + 1];
      A3.x = xs[m + 48][kk];  A3.y = xs[m + 48][kk + 1];
      acc0 = __builtin_amdgcn_wmma_f32_16x16x4_f32(false, A0, false, B,
                                                   (short)0, acc0, false, false);
      acc1 = __builtin_amdgcn_wmma_f32_16x16x4_f32(false, A1, false, B,
                                                   (short)0, acc1, false, false);
      acc2 = __builtin_amdgcn_wmma_f32_16x16x4_f32(false, A2, false, B,
                                                   (short)0, acc2, false, false);
      acc3 = __builtin_amdgcn_wmma_f32_16x16x4_f32(false, A3, false, B,
                                                   (short)0, acc3, false, false);
    }
    // C/D layout: VGPR r -> lanes 0-15: M=r, N=lane ; lanes 16-31: M=r+8.
    const int mbase = row0 + (hi << 3);
#pragma unroll
    for (int r = 0; r < 8; ++r) {
      const int r0 = mbase + r;
      if (r0 < nrows)      H[(size_t)r0 * HC + c0 + m]        = acc0[r];
      if (r0 + 16 < nrows) H[(size_t)(r0 + 16) * HC + c0 + m] = acc1[r];
      if (r0 + 32 < nrows) H[(size_t)(r0 + 32) * HC + c0 + m] = acc2[r];
      if (r0 + 48 < nrows) H[(size_t)(r0 + 48) * HC + c0 + m] = acc3[r];
    }
  }
}

// ---------------------------------------------------------------------------
// a_src[n,h] = sum_c h[n,h*64+c]*att_src[h*64+c]  (same for a_dst).
// One wave per node; lane owns 8 consecutive channels (all in one head);
// butterfly-reduce within each 8-lane head group.
// ---------------------------------------------------------------------------
__global__ __launch_bounds__(256) void attn_scores(
    const float* __restrict__ H, const float* __restrict__ att_src,
    const float* __restrict__ att_dst, float* __restrict__ a_src,
    float* __restrict__ a_dst, int nnodes) {
  const int wave = threadIdx.x >> 5;
  const int lane = threadIdx.x & 31;
  const int node = blockIdx.x * 8 + wave;
  if (node >= nnodes) return;
  float ps = 0.f, pd = 0.f;
  const float* hp = H + (size_t)node * HC + lane * 8;
#pragma unroll
  for (int j = 0; j < 8; ++j) {
    const float hv = hp[j];
    ps += hv * att_src[lane * 8 + j];
    pd += hv * att_dst[lane * 8 + j];
  }
#pragma unroll
  for (int s = 1; s < 8; s <<= 1) {
    ps += __shfl_xor(ps, s, 32);
    pd += __shfl_xor(pd, s, 32);
  }
  if ((lane & 7) == 0) {
    a_src[node * HEADS + (lane >> 3)] = ps;
    a_dst[node * HEADS + (lane >> 3)] = pd;
  }
}

// ---------------------------------------------------------------------------
// acc = 0, m = -inf, denom = 0 (grid-stride).
// ---------------------------------------------------------------------------
__global__ void init_bufs(float* __restrict__ acc, float* __restrict__ m,
                          float* __restrict__ denom, int accN, int nh) {
  const int stride = gridDim.x * blockDim.x;
  for (int j = blockIdx.x * blockDim.x + threadIdx.x; j < accN; j += stride)
    acc[j] = 0.f;
  for (int j = blockIdx.x * blockDim.x + threadIdx.x; j < nh; j += stride) {
    m[j] = -__builtin_inff();
    denom[j] = 0.f;
  }
}

__device__ __forceinline__ float leaky(float v) {
  return v > 0.f ? v : SLOPE * v;
}

// Monotonic-bit-pattern float atomic max (m initialized to -inf).
__device__ __forceinline__ void atomicMaxF(float* addr, float val) {
  if (val >= 0.f)
    atomicMax((int*)addr, __float_as_int(val));
  else
    atomicMin((unsigned int*)addr, __float_as_uint(val));
}

// ---------------------------------------------------------------------------
// Pass 1: segment max of e over dst. Thread per (edge, head).
// ---------------------------------------------------------------------------
__global__ void edge_max(const int* __restrict__ src, const int* __restrict__ dst,
                         const float* __restrict__ a_src,
                         const float* __restrict__ a_dst,
                         float* __restrict__ m, int nE) {
  const int t = blockIdx.x * blockDim.x + threadIdx.x;
  const int e = t >> 2, h = t & 3;
  if (e >= nE) return;
  const int s = src[e], d = dst[e];
  const float v = leaky(a_src[s * HEADS + h] + a_dst[d * HEADS + h]);
  atomicMaxF(&m[d * HEADS + h], v);
}

// ---------------------------------------------------------------------------
// Pass 2: denom[dst,h] += exp(e - m[dst,h]). e recomputed (cheaper than
// storing 850K x 4 floats; a_src/a_dst are L2-resident).
// ---------------------------------------------------------------------------
__global__ void edge_sum(const int* __restrict__ src, const int* __restrict__ dst,
                         const float* __restrict__ a_src,
                         const float* __restrict__ a_dst,
                         const float* __restrict__ m,
                         float* __restrict__ denom, int nE) {
  const int t = blockIdx.x * blockDim.x + threadIdx.x;
  const int e = t >> 2, h = t & 3;
  if (e >= nE) return;
  const int s = src[e], d = dst[e];
  const float v = leaky(a_src[s * HEADS + h] + a_dst[d * HEADS + h]);
  atomicAdd(&denom[d * HEADS + h], __expf(v - m[d * HEADS + h]));
}

// ---------------------------------------------------------------------------
// Pass 3: acc[dst] += h[src] * alpha. One wave per edge; lane owns 8 channels
// (one head). h and acc are L2-resident (51.2 MB each, 192 MB L2), so the
// gathers and f32 atomic adds ride on L2 bandwidth, not HBM.
// ---------------------------------------------------------------------------
__global__ __launch_bounds__(256) void edge_msg(
    const int* __restrict__ src, const int* __restrict__ dst,
    const float* __restrict__ a_src, const float* __restrict__ a_dst,
    const float* __restrict__ m, const float* __restrict__ denom,
    const float* __restrict__ H, float* __restrict__ acc, int nE) {
  const int wave = threadIdx.x >> 5;
  const int lane = threadIdx.x & 31;
  const int e = blockIdx.x * 8 + wave;
  if (e >= nE) return;
  const int s = src[e], d = dst[e];
  const int h = lane >> 3;
  const float ev    = leaky(a_src[s * HEADS + h] + a_dst[d * HEADS + h]);
  const float ex    = __expf(ev - m[d * HEADS + h]);
  const float alpha = ex / (denom[d * HEADS + h] + 1e-16f);
  const float4* hs = (const float4*)(H + (size_t)s * HC + lane * 8);
  const float4 v0 = hs[0], v1 = hs[1];
  float* op = acc + (size_t)d * HC + lane * 8;
  atomicAdd(op + 0, v0.x * alpha);
  atomicAdd(op + 1, v0.y * alpha);
  atomicAdd(op + 2, v0.z * alpha);
  atomicAdd(op + 3, v0.w * alpha);
  atomicAdd(op + 4, v1.x * alpha);
  atomicAdd(op + 5, v1.y * alpha);
  atomicAdd(op + 6, v1.z * alpha);
  atomicAdd(op + 7, v1.w * alpha);
}

// ---------------------------------------------------------------------------
// out = relu(acc + bias)  (in place for layers 0/1, into d_out for layer 2).
// ---------------------------------------------------------------------------
__global__ void bias_relu(const float* __restrict__ acc,
                          const float* __restrict__ bias,
                          float* __restrict__ out, int total) {
  const int i = blockIdx.x * blockDim.x + threadIdx.x;
  if (i >= total) return;
  const float v = acc[i] + bias[i & (HC - 1)];
  out[i] = v > 0.f ? v : 0.f;
}

extern "C" void kernel_launch(void* const* d_in, const int* in_sizes, int n_in,
                              void* d_out, int out_size, void* d_ws,
                              size_t ws_size, hipStream_t stream) {
  const float* x0 = (const float*)d_in[0];
  const int* ei = (const int*)d_in[1];
  const int nN = in_sizes[0] / 128;  // IN0 = 128
  const int nE = in_sizes[1] / 2;    // edge_index is [2, E]
  const int* srcp = ei;
  const int* dstp = ei + nE;

  // Workspace layout (floats): H [N*256] | acc/x [N*256] | a_src | a_dst | m | denom
  float* Hbuf = (float*)d_ws;
  float* Xbuf = Hbuf + (size_t)nN * HC;
  float* asrc = Xbuf + (size_t)nN * HC;
  float* adst = asrc + (size_t)nN * HEADS;
  float* mbuf = adst + (size_t)nN * HEADS;
  float* dbuf = mbuf + (size_t)nN * HEADS;

  const float* Wl[3] = {(const float*)d_in[2], (const float*)d_in[6],
                        (const float*)d_in[10]};
  const float* As[3] = {(const float*)d_in[3], (const float*)d_in[7],
                        (const float*)d_in[11]};
  const float* Ad[3] = {(const float*)d_in[4], (const float*)d_in[8],
                        (const float*)d_in[12]};
  const float* Bi[3] = {(const float*)d_in[5], (const float*)d_in[9],
                        (const float*)d_in[13]};

  const int rowStrips = (nN + 63) / 64;
  for (int l = 0; l < 3; ++l) {
    const float* xin = (l == 0) ? x0 : Xbuf;
    if (l == 0)
      gemm_wmma_f32<128><<<rowStrips, 256, 0, stream>>>(xin, Wl[l], Hbuf, nN);
    else
      gemm_wmma_f32<256><<<rowStrips, 256, 0, stream>>>(xin, Wl[l], Hbuf, nN);

    attn_scores<<<(nN + 7) / 8, 256, 0, stream>>>(Hbuf, As[l], Ad[l], asrc,
                                                  adst, nN);
    init_bufs<<<2048, 256, 0, stream>>>(Xbuf, mbuf, dbuf, nN * HC, nN * HEADS);
    edge_max<<<(nE * 4 + 255) / 256, 256, 0, stream>>>(srcp, dstp, asrc, adst,
                                                       mbuf, nE);
    edge_sum<<<(nE * 4 + 255) / 256, 256, 0, stream>>>(srcp, dstp, asrc, adst,
                                                       mbuf, dbuf, nE);
    edge_msg<<<(nE + 7) / 8, 256, 0, stream>>>(srcp, dstp, asrc, adst, mbuf,
                                               dbuf, Hbuf, Xbuf, nE);
    float* outp = (l == 2) ? (float*)d_out : Xbuf;
    bias_relu<<<(nN * HC + 255) / 256, 256, 0, stream>>>(Xbuf, Bi[l], outp,
                                                         nN * HC);
  }
}